// SparseConvolutionDownsample_25280177504761
// MI455X (gfx1250) — compile-verified
//
#include <hip/hip_runtime.h>

typedef float v2f __attribute__((ext_vector_type(2)));
typedef float v8f __attribute__((ext_vector_type(8)));

#define N_IN   2000000
#define N_OUT  500000
#define K_OFF  4
#define RB     500000
#define C_IN   32
#define C_OUT  64
#define BN_EPS 1e-4f
#define TILES_PER_OFF (RB / 16)   // 31250

// ---------------------------------------------------------------------------
// Kernel 1: zero the output accumulator and the stats scratch (ws[0..127])
// ---------------------------------------------------------------------------
__global__ __launch_bounds__(256)
void zero_acc_kernel(float4* __restrict__ out4, long n4, float* __restrict__ ws) {
    long i = (long)blockIdx.x * blockDim.x + threadIdx.x;
    const long stride = (long)gridDim.x * blockDim.x;
    const float4 z = make_float4(0.f, 0.f, 0.f, 0.f);
    for (; i < n4; i += stride) out4[i] = z;
    if (blockIdx.x == 0 && threadIdx.x < 128) ws[threadIdx.x] = 0.f;
}

// ---------------------------------------------------------------------------
// Kernel 2: rulebook gather -> WMMA fp32 GEMM -> scatter atomic-add
//   blockIdx.y = filter offset k, 8 waves per block, each wave owns 16-row
//   rulebook tiles. Weight[k] staged in LDS, B tiles hoisted to VGPRs.
// ---------------------------------------------------------------------------
__global__ __launch_bounds__(256)
void spconv_gemm_scatter_kernel(const float* __restrict__ x,
                                const float* __restrict__ w,
                                const int*   __restrict__ in_idx,
                                const int*   __restrict__ out_idx,
                                float*       __restrict__ out) {
    __shared__ float lw[C_IN * C_OUT];   // 8 KB: weight for this offset

    const int k = blockIdx.y;
    for (int i = threadIdx.x; i < C_IN * C_OUT; i += 256)
        lw[i] = w[k * C_IN * C_OUT + i];
    __syncthreads();

    const int lane = threadIdx.x & 31;
    const int wv   = threadIdx.x >> 5;      // wave in block (0..7)
    const int l15  = lane & 15;
    const int hi   = lane >> 4;             // half-wave (0/1)

    // B-matrix register tiles: b[n][kk] covers rows 4*kk + 2*hi + {0,1},
    // column n*16 + l15 of the 32x64 weight slab (4x16 f32 B layout).
    v2f b[4][8];
#pragma unroll
    for (int n = 0; n < 4; ++n) {
#pragma unroll
        for (int kk = 0; kk < 8; ++kk) {
            const int row = kk * 4 + hi * 2;
            const int col = n * 16 + l15;
            v2f t;
            t.x = lw[row * C_OUT + col];
            t.y = lw[(row + 1) * C_OUT + col];
            b[n][kk] = t;
        }
    }

    const long base_rb = (long)k * RB;

    for (int tile = blockIdx.x * 8 + wv; tile < TILES_PER_OFF; tile += gridDim.x * 8) {
        const long tb = base_rb + (long)tile * 16;

        // A-matrix gather: lane's matrix row m = l15, K pair starts at 2*hi.
        const int im = in_idx[tb + l15];
        const float* xrow = x + (long)im * C_IN + hi * 2;

        v8f acc0 = {}, acc1 = {}, acc2 = {}, acc3 = {};
#pragma unroll
        for (int kk = 0; kk < 8; ++kk) {
            const v2f a = *(const v2f*)(xrow + kk * 4);   // 8B-aligned b64 load
            acc0 = __builtin_amdgcn_wmma_f32_16x16x4_f32(false, a, false, b[0][kk],
                                                         (short)0, acc0, false, false);
            acc1 = __builtin_amdgcn_wmma_f32_16x16x4_f32(false, a, false, b[1][kk],
                                                         (short)0, acc1, false, false);
            acc2 = __builtin_amdgcn_wmma_f32_16x16x4_f32(false, a, false, b[2][kk],
                                                         (short)0, acc2, false, false);
            acc3 = __builtin_amdgcn_wmma_f32_16x16x4_f32(false, a, false, b[3][kk],
                                                         (short)0, acc3, false, false);
        }

        // Scatter-add. C layout: lane 0-15 VGPR v -> M=v, N=lane;
        //                        lane 16-31 VGPR v -> M=8+v, N=lane-16.
#pragma unroll
        for (int v = 0; v < 8; ++v) {
            const int m  = hi * 8 + v;
            const int oi = out_idx[tb + m];
            float* orow = out + (long)oi * C_OUT + l15;
            atomicAdd(orow +  0, acc0[v]);
            atomicAdd(orow + 16, acc1[v]);
            atomicAdd(orow + 32, acc2[v]);
            atomicAdd(orow + 48, acc3[v]);
        }
    }
}

// ---------------------------------------------------------------------------
// Kernel 3: per-channel sum / sum-of-squares -> ws[0..63], ws[64..127]
// ---------------------------------------------------------------------------
__global__ __launch_bounds__(256)
void bn_stats_kernel(const float* __restrict__ out, float* __restrict__ ws) {
    __shared__ float ss[256];
    __shared__ float sq[256];
    const int c  = threadIdx.x & 63;
    const int rg = threadIdx.x >> 6;    // row group 0..3
    float s = 0.f, q = 0.f;
    for (long r = (long)blockIdx.x * 4 + rg; r < N_OUT; r += (long)gridDim.x * 4) {
        const float v = out[r * C_OUT + c];
        s += v;
        q += v * v;
    }
    ss[threadIdx.x] = s;
    sq[threadIdx.x] = q;
    __syncthreads();
    if (threadIdx.x < 64) {
        const float ts = ss[c] + ss[c + 64] + ss[c + 128] + ss[c + 192];
        const float tq = sq[c] + sq[c + 64] + sq[c + 128] + sq[c + 192];
        atomicAdd(&ws[c], ts);
        atomicAdd(&ws[64 + c], tq);
    }
}

// ---------------------------------------------------------------------------
// Kernel 4: BatchNorm (batch stats) + ReLU, in place on d_out
// ---------------------------------------------------------------------------
__global__ __launch_bounds__(256)
void bn_relu_kernel(float* __restrict__ out, const float* __restrict__ ws,
                    const float* __restrict__ gamma, const float* __restrict__ beta) {
    const float invN = 1.0f / (float)N_OUT;
    long i = (long)blockIdx.x * blockDim.x + threadIdx.x;
    const long stride = (long)gridDim.x * blockDim.x;
    const long total  = (long)N_OUT * C_OUT;
    for (; i < total; i += stride) {
        const int   c    = (int)(i & 63);
        const float mean = ws[c] * invN;
        const float var  = ws[64 + c] * invN - mean * mean;
        const float sc   = gamma[c] * rsqrtf(var + BN_EPS);
        const float y    = (out[i] - mean) * sc + beta[c];
        out[i] = fmaxf(y, 0.f);
    }
}

// ---------------------------------------------------------------------------
extern "C" void kernel_launch(void* const* d_in, const int* in_sizes, int n_in,
                              void* d_out, int out_size, void* d_ws, size_t ws_size,
                              hipStream_t stream) {
    const float* x       = (const float*)d_in[0];
    const float* w       = (const float*)d_in[1];
    const float* gamma   = (const float*)d_in[2];
    const float* beta    = (const float*)d_in[3];
    const int*   in_idx  = (const int*)d_in[4];
    const int*   out_idx = (const int*)d_in[5];
    float* out = (float*)d_out;
    float* ws  = (float*)d_ws;

    const long n4 = (long)N_OUT * C_OUT / 4;
    zero_acc_kernel<<<2048, 256, 0, stream>>>((float4*)out, n4, ws);

    dim3 g2(512, K_OFF);
    spconv_gemm_scatter_kernel<<<g2, 256, 0, stream>>>(x, w, in_idx, out_idx, out);

    bn_stats_kernel<<<1024, 256, 0, stream>>>(out, ws);

    bn_relu_kernel<<<2048, 256, 0, stream>>>(out, ws, gamma, beta);
}